// GATNet_7069516169315
// MI455X (gfx1250) — compile-verified
//
#include <hip/hip_runtime.h>
#include <math.h>

typedef __attribute__((ext_vector_type(2))) float v2f;
typedef __attribute__((ext_vector_type(8))) float v8f;

namespace {
constexpr int N_  = 50000;    // nodes
constexpr int ET_ = 550000;   // E + N self loops
constexpr int D_  = 128;      // feature dim
constexpr int H_  = 2;        // heads
constexpr int DH_ = 64;       // dim per head
constexpr int NT_ = 100000;   // triples
constexpr int K_  = 3;        // negatives per triple
constexpr int NR_ = 20000;    // rules
constexpr int NS_ = 10000;    // sampled nodes
constexpr float ALPHA_ = 0.2f;
}

// ---------- helpers ----------
__device__ __forceinline__ float wave_sum(float v) {
#pragma unroll
  for (int o = 16; o > 0; o >>= 1) v += __shfl_down(v, o, 32);
  return v;
}

// monotone float<->uint key (valid for all non-NaN floats)
__device__ __forceinline__ unsigned f2ord(float f) {
  int i = __float_as_int(f);
  return (unsigned)(i ^ ((i >> 31) | (int)0x80000000));
}
__device__ __forceinline__ float ord2f(unsigned u) {
  int i = (u & 0x80000000u) ? (int)(u ^ 0x80000000u) : ~(int)u;
  return __int_as_float(i);
}

// ---------- GEMM: Hout[n, h*64+k] = sum_d X[n,d] * Ws[l,h,d,k] (fp32 WMMA) ----------
// One wave computes a 16x16 tile; K loop = 32 x v_wmma_f32_16x16x4_f32.
__global__ __launch_bounds__(32) void gemm_wmma_k(const float* __restrict__ X,
                                                  const float* __restrict__ Wl,
                                                  float* __restrict__ Hout) {
  const int lane = threadIdx.x;
  const int half = lane >> 4;        // 0: K pair {0,1}, 1: K pair {2,3}
  const int l16  = lane & 15;
  const int rowA = (blockIdx.x << 4) + l16;   // A-matrix row (M)
  const int col  = (blockIdx.y << 4) + l16;   // B/D column (N)
  const int head = col >> 6;
  const int kcol = col & 63;
  const float* __restrict__ xa = X + rowA * D_;
  const float* __restrict__ wb = Wl + head * (D_ * DH_) + kcol;  // stride DH_ over d
  v8f acc = {};
  for (int k0 = 0; k0 < D_; k0 += 4) {
    const int ka = k0 + (half << 1);
    v2f a, b;
    a.x = xa[ka];
    a.y = xa[ka + 1];
    b.x = wb[ka * DH_];
    b.y = wb[(ka + 1) * DH_];
    acc = __builtin_amdgcn_wmma_f32_16x16x4_f32(false, a, false, b,
                                                (short)0, acc, false, false);
  }
  const int mBase = (blockIdx.x << 4) + (half << 3);
#pragma unroll
  for (int r = 0; r < 8; ++r)
    Hout[(mBase + r) * D_ + col] = acc[r];
}

// ---------- per-(node,head) attention coefficients ----------
__global__ void attn_coef_k(const float* __restrict__ Hf, const float* __restrict__ Al,
                            float* __restrict__ asrc, float* __restrict__ adst) {
  int i = blockIdx.x * blockDim.x + threadIdx.x;      // over N_*H_
  if (i >= N_ * H_) return;
  int node = i >> 1, h = i & 1;
  const float* hv = Hf + node * D_ + h * DH_;
  const float* as = Al + h * (2 * DH_);
  const float* ad = as + DH_;
  float s1 = 0.f, s2 = 0.f;
#pragma unroll 4
  for (int k = 0; k < DH_; ++k) { float v = hv[k]; s1 += v * as[k]; s2 += v * ad[k]; }
  asrc[i] = s1; adst[i] = s2;
}

// ---------- edge pass 1: segment max of leaky_relu(a_src[s]+a_dst[d]) ----------
__global__ void edge_max_k(const int* __restrict__ src, const int* __restrict__ dst,
                           const float* __restrict__ asrc, const float* __restrict__ adst,
                           unsigned* __restrict__ mkeys) {
  int i = blockIdx.x * blockDim.x + threadIdx.x;      // over ET_*H_
  if (i >= ET_ * H_) return;
  int e = i >> 1, h = i & 1;
  int s = src[e], d = dst[e];
  float v = asrc[s * H_ + h] + adst[d * H_ + h];
  v = v >= 0.f ? v : ALPHA_ * v;
  atomicMax(&mkeys[d * H_ + h], f2ord(v));
}

// ---------- edge pass 2: ex = exp(e - m[dst]); den[dst] += ex ----------
__global__ void edge_expsum_k(const int* __restrict__ src, const int* __restrict__ dst,
                              const float* __restrict__ asrc, const float* __restrict__ adst,
                              const unsigned* __restrict__ mkeys,
                              float* __restrict__ exb, float* __restrict__ den) {
  int i = blockIdx.x * blockDim.x + threadIdx.x;      // over ET_*H_
  if (i >= ET_ * H_) return;
  int e = i >> 1, h = i & 1;
  int s = src[e], d = dst[e];
  float v = asrc[s * H_ + h] + adst[d * H_ + h];
  v = v >= 0.f ? v : ALPHA_ * v;
  float x = expf(v - ord2f(mkeys[d * H_ + h]));
  exb[i] = x;
  atomicAdd(&den[d * H_ + h], x);
}

// ---------- edge pass 3: out[dst] += (ex/den[dst]) * h[src], 4 floats/thread ----------
__global__ void edge_aggr_k(const int* __restrict__ src, const int* __restrict__ dst,
                            const float* __restrict__ exb, const float* __restrict__ den,
                            const float* __restrict__ Hf, float* __restrict__ aggr) {
  long i = (long)blockIdx.x * blockDim.x + threadIdx.x;   // over ET_*32
  if (i >= (long)ET_ * 32) return;
  int e = (int)(i >> 5);
  int q = (int)(i & 31);
  int c0 = q << 2;               // feature chunk start (0..124)
  int h = c0 >> 6;               // head of this chunk
  int s = src[e], d = dst[e];
  float coeff = exb[e * H_ + h] / den[d * H_ + h];
  const float4 hv = *(const float4*)(Hf + s * D_ + c0);
  float* o = aggr + d * D_ + c0;
  atomicAdd(o + 0, coeff * hv.x);
  atomicAdd(o + 1, coeff * hv.y);
  atomicAdd(o + 2, coeff * hv.z);
  atomicAdd(o + 3, coeff * hv.w);
}

// ---------- elementwise ELU (in place) ----------
__global__ void elu_k(float* __restrict__ p, long n) {
  long i = (long)blockIdx.x * blockDim.x + threadIdx.x;
  if (i >= n) return;
  float x = p[i];
  p[i] = x > 0.f ? x : (expf(x) - 1.f);
}

// ---------- TransE: sc[tj, c] = out[h]+rel[r]-out[t], 4 floats/thread ----------
__global__ void transe_k(const float* __restrict__ outg, const float* __restrict__ rel,
                         const int* __restrict__ hI, const int* __restrict__ tI,
                         const int* __restrict__ rI, float* __restrict__ sc) {
  long i = (long)blockIdx.x * blockDim.x + threadIdx.x;   // over NT_*K_*32
  if (i >= (long)NT_ * K_ * 32) return;
  int tj = (int)(i >> 5);
  int c0 = (int)(i & 31) << 2;
  int h = hI[tj], t = tI[tj], r = rI[tj];
  const float4 a = *(const float4*)(outg + (long)h * D_ + c0);
  const float4 b = *(const float4*)(rel  + (long)r * D_ + c0);
  const float4 c = *(const float4*)(outg + (long)t * D_ + c0);
  float4 o;
  o.x = a.x + b.x - c.x; o.y = a.y + b.y - c.y;
  o.z = a.z + b.z - c.z; o.w = a.w + b.w - c.w;
  *(float4*)(sc + (long)tj * D_ + c0) = o;
}

// ---------- truth value of sc[:,0,:]: one wave per triple ----------
__global__ void truth_k(const float* __restrict__ sc, float* __restrict__ tv) {
  long gid = (long)blockIdx.x * blockDim.x + threadIdx.x;
  if (gid == 0) tv[NT_] = 1.0f;                 // the appended "ones" row
  int t = (int)(gid >> 5);
  int lane = (int)(gid & 31);
  if (t >= NT_) return;
  const float4 v = *(const float4*)(sc + (long)t * (K_ * D_) + lane * 4);
  float s  = v.x + v.y + v.z + v.w;
  float ss = v.x * v.x + v.y * v.y + v.z * v.z + v.w * v.w;
  s  = wave_sum(s);
  ss = wave_sum(ss);
  if (lane == 0) {
    float nrm = sqrtf(ss);
    nrm = fmaxf(nrm, 1e-12f);
    tv[t] = 1.0f - s / (nrm * (3.0f * sqrtf((float)D_)));
  }
}

// ---------- rules: one wave per rule ----------
__global__ void rule_k(const float* __restrict__ ent, const float* __restrict__ rel,
                       const int* __restrict__ rh, const int* __restrict__ rt,
                       const int* __restrict__ rr, const int* __restrict__ prem,
                       const float* __restrict__ tv, float* __restrict__ outR) {
  long gid = (long)blockIdx.x * blockDim.x + threadIdx.x;
  int w = (int)(gid >> 5);
  int lane = (int)(gid & 31);
  if (w >= NR_) return;
  int h = rh[w], t = rt[w], r = rr[w];
  const float4 a = *(const float4*)(ent + (long)h * D_ + lane * 4);
  const float4 b = *(const float4*)(rel + (long)r * D_ + lane * 4);
  const float4 c = *(const float4*)(ent + (long)t * D_ + lane * 4);
  float4 v;
  v.x = a.x + b.x - c.x; v.y = a.y + b.y - c.y;
  v.z = a.z + b.z - c.z; v.w = a.w + b.w - c.w;
  float s  = v.x + v.y + v.z + v.w;
  float ss = v.x * v.x + v.y * v.y + v.z * v.z + v.w * v.w;
  s  = wave_sum(s);
  ss = wave_sum(ss);
  if (lane == 0) {
    float nrm = fmaxf(sqrtf(ss), 1e-12f);
    float rs = 1.0f - s / (nrm * (3.0f * sqrtf((float)D_)));
    float f1 = tv[prem[w * 2 + 0]];
    float f2 = tv[prem[w * 2 + 1]];
    outR[w] = 1.0f + f1 * f2 * (rs - 1.0f);
  }
}

// ---------- gather out[data] ----------
__global__ void gather_k(const float* __restrict__ outg, const int* __restrict__ idx,
                         float* __restrict__ o) {
  long i = (long)blockIdx.x * blockDim.x + threadIdx.x;   // over NS_*D_
  if (i >= (long)NS_ * D_) return;
  int row = (int)(i >> 7);
  int d = (int)(i & 127);
  o[i] = outg[(long)idx[row] * D_ + d];
}

// ---------- host orchestration ----------
extern "C" void kernel_launch(void* const* d_in, const int* in_sizes, int n_in,
                              void* d_out, int out_size, void* d_ws, size_t ws_size,
                              hipStream_t stream) {
  (void)in_sizes; (void)n_in; (void)out_size; (void)ws_size;
  const float* Ws = (const float*)d_in[4];   // (L,H,D,dh)
  const float* As = (const float*)d_in[5];   // (L,H,2*dh)

  // workspace layout (floats, all 4-byte aligned; d_ws is 256B aligned)
  float* ws    = (float*)d_ws;
  float* hfeat = ws;                         // N*D
  float* tmp   = hfeat + (long)N_ * D_;      // N*D
  float* osr   = tmp   + (long)N_ * D_;      // N*D
  float* otg   = osr   + (long)N_ * D_;      // N*D
  float* asrc  = otg   + (long)N_ * D_;      // N*H
  float* adst  = asrc  + (long)N_ * H_;      // N*H
  unsigned* mkeys = (unsigned*)(adst + (long)N_ * H_);   // N*H
  float* den   = (float*)(mkeys + (long)N_ * H_);        // N*H
  float* exb   = den   + (long)N_ * H_;      // ET*H
  float* tv    = exb   + (long)ET_ * H_;     // NT+1

  float* out = (float*)d_out;
  const long OFF_GTG    = (long)NS_ * D_;
  const long OFF_TRANSE = 2L * NS_ * D_;
  const long OFF_RULE   = OFF_TRANSE + 2L * NT_ * K_ * D_;

  const dim3 gGemm(N_ / 16, D_ / 16);
  const int BT = 256;
  const int gNH   = (N_ * H_ + BT - 1) / BT;
  const int gEH   = (ET_ * H_ + BT - 1) / BT;
  const long nAgg = (long)ET_ * 32;
  const int gAgg  = (int)((nAgg + BT - 1) / BT);
  const long nND  = (long)N_ * D_;
  const int gND   = (int)((nND + BT - 1) / BT);

  for (int g = 0; g < 2; ++g) {
    const float* ent = (const float*)d_in[0 + g];
    const int* src   = (const int*)d_in[6 + 2 * g];
    const int* dst   = (const int*)d_in[7 + 2 * g];
    float* outg = (g == 0) ? osr : otg;

    for (int l = 0; l < 2; ++l) {
      const float* x  = (l == 0) ? ent : tmp;
      float* aggr     = (l == 0) ? tmp : outg;
      const float* Wl = Ws + (long)l * H_ * D_ * DH_;
      const float* Al = As + (long)l * H_ * 2 * DH_;

      gemm_wmma_k<<<gGemm, 32, 0, stream>>>(x, Wl, hfeat);
      attn_coef_k<<<gNH, BT, 0, stream>>>(hfeat, Al, asrc, adst);
      hipMemsetAsync(mkeys, 0, (size_t)N_ * H_ * sizeof(unsigned), stream);
      hipMemsetAsync(den,   0, (size_t)N_ * H_ * sizeof(float), stream);
      hipMemsetAsync(aggr,  0, (size_t)N_ * D_ * sizeof(float), stream);
      edge_max_k<<<gEH, BT, 0, stream>>>(src, dst, asrc, adst, mkeys);
      edge_expsum_k<<<gEH, BT, 0, stream>>>(src, dst, asrc, adst, mkeys, exb, den);
      edge_aggr_k<<<gAgg, BT, 0, stream>>>(src, dst, exb, den, hfeat, aggr);
      if (l == 0) elu_k<<<gND, BT, 0, stream>>>(tmp, nND);
    }
  }

  // TransE scores, truth values, rules (sequential per graph; tv reused)
  const long nTE = (long)NT_ * K_ * 32;
  const int gTE  = (int)((nTE + BT - 1) / BT);
  const int gTV  = (int)(((long)NT_ * 32 + BT - 1) / BT);
  const int gRU  = (int)(((long)NR_ * 32 + BT - 1) / BT);
  const int gGA  = (int)(((long)NS_ * D_ + BT - 1) / BT);

  for (int g = 0; g < 2; ++g) {
    const float* ent = (const float*)d_in[0 + g];
    const float* rel = (const float*)d_in[2 + g];
    const float* outg = (g == 0) ? osr : otg;
    float* sc = out + OFF_TRANSE + (long)g * NT_ * K_ * D_;

    transe_k<<<gTE, BT, 0, stream>>>(outg, rel,
        (const int*)d_in[12 + 3 * g], (const int*)d_in[13 + 3 * g],
        (const int*)d_in[14 + 3 * g], sc);
    truth_k<<<gTV, BT, 0, stream>>>(sc, tv);
    rule_k<<<gRU, BT, 0, stream>>>(ent, rel,
        (const int*)d_in[18 + 4 * g], (const int*)d_in[19 + 4 * g],
        (const int*)d_in[20 + 4 * g], (const int*)d_in[21 + 4 * g],
        tv, out + OFF_RULE + (long)g * NR_);
  }

  gather_k<<<gGA, BT, 0, stream>>>(osr, (const int*)d_in[10], out);
  gather_k<<<gGA, BT, 0, stream>>>(otg, (const int*)d_in[11], out + OFF_GTG);
}